// ChamferDistance_14714557956155
// MI455X (gfx1250) — compile-verified
//
#include <hip/hip_runtime.h>

typedef __attribute__((ext_vector_type(2))) float v2f;
typedef __attribute__((ext_vector_type(8))) float v8f;

__global__ void chamfer_zero_kernel(float* out) { out[0] = 0.0f; }

// One wave handles a 16-row tile of A, streams all of B in 16-column tiles.
// Squared distance comes entirely out of the matrix unit:
//   A' = [-2x, -2y, -2z, 1]  (16x4),  B' = [x, y, z, ||b||^2] (4x16),
//   C  = ||a||^2 broadcast per row
//   D  = A' x B' + C = ||a||^2 - 2 a.b + ||b||^2  (one v_wmma_f32_16x16x4_f32 per tile)
__global__ __launch_bounds__(256) void chamfer_dir_kernel(
    const float* __restrict__ A, const float* __restrict__ B,
    float* __restrict__ out, int n)
{
    const int lane = threadIdx.x & 31;
    const int wave = blockIdx.x * (blockDim.x >> 5) + (threadIdx.x >> 5);
    const int rowBase = wave * 16;
    const int half = lane >> 4;   // 0: lanes 0-15, 1: lanes 16-31
    const int l15  = lane & 15;

    // Each lane loads A point for row rowBase + (lane&15) (both halves load it).
    const float* ap = A + 3 * (rowBase + l15);
    const float ax = ap[0], ay = ap[1], az = ap[2];
    const float na = ax * ax + ay * ay + az * az;

    // A operand (16x4): lanes 0-15: (K0,K1) = (-2x,-2y); lanes 16-31: (K2,K3) = (-2z, 1)
    v2f a;
    a.x = (half == 0) ? (-2.0f * ax) : (-2.0f * az);
    a.y = (half == 0) ? (-2.0f * ay) : 1.0f;

    // C matrix: C_v[lanes 0-15] = ||a_(v)||^2, C_v[lanes 16-31] = ||a_(v+8)||^2.
    v8f cna;
#pragma unroll
    for (int v = 0; v < 8; ++v)
        cna[v] = __shfl(na, v + (half << 3), 32);

    v8f mn;
#pragma unroll
    for (int v = 0; v < 8; ++v) mn[v] = 3.4e38f;

    for (int col = 0; col < n; col += 64) {
#pragma unroll
        for (int u = 0; u < 4; ++u) {
            const float* bq = B + 3 * (col + u * 16 + l15);
            const float bx = bq[0], by = bq[1], bz = bq[2];
            const float nb = bx * bx + by * by + bz * bz;

            // B operand (4x16): lanes 0-15: (K0,K1)=(x,y); lanes 16-31: (K2,K3)=(z,||b||^2)
            v2f b;
            b.x = (half == 0) ? bx : bz;
            b.y = (half == 0) ? by : nb;

            // D = full squared-distance tile
            v8f d = __builtin_amdgcn_wmma_f32_16x16x4_f32(
                false, a, false, b, (short)0, cna, false, false);

#pragma unroll
            for (int v = 0; v < 8; ++v)
                mn[v] = fminf(mn[v], d[v]);
        }
    }

    // Columns are striped (col % 16 == lane&15): min across the 16 lanes of each half.
#pragma unroll
    for (int v = 0; v < 8; ++v) {
        float m = mn[v];
        m = fminf(m, __shfl_xor(m, 1, 32));
        m = fminf(m, __shfl_xor(m, 2, 32));
        m = fminf(m, __shfl_xor(m, 4, 32));
        m = fminf(m, __shfl_xor(m, 8, 32));
        mn[v] = m;
    }

    // Sum the 8 row-mins held by this half; lanes 0 and 16 carry rows 0-7 / 8-15.
    float s = 0.0f;
#pragma unroll
    for (int v = 0; v < 8; ++v) s += mn[v];

    if (l15 == 0)
        atomicAdd(out, s * (1.0f / (float)n));
}

extern "C" void kernel_launch(void* const* d_in, const int* in_sizes, int n_in,
                              void* d_out, int out_size, void* d_ws, size_t ws_size,
                              hipStream_t stream) {
    const float* gt  = (const float*)d_in[0];
    const float* gen = (const float*)d_in[1];
    float* out = (float*)d_out;

    const int n = in_sizes[0] / 3;          // 16384 points, divisible by 64
    const int wavesPerBlock = 8;            // 256 threads = 8 wave32
    const int rowsPerBlock = 16 * wavesPerBlock;
    const int blocks = n / rowsPerBlock;    // 128

    chamfer_zero_kernel<<<1, 1, 0, stream>>>(out);
    // gt -> gen direction
    chamfer_dir_kernel<<<blocks, 256, 0, stream>>>(gt, gen, out, n);
    // gen -> gt direction
    chamfer_dir_kernel<<<blocks, 256, 0, stream>>>(gen, gt, out, n);
}